// LinearKalmanFilter_17609365914484
// MI455X (gfx1250) — compile-verified
//
#include <hip/hip_runtime.h>

typedef float v2f __attribute__((ext_vector_type(2)));
typedef float v8f __attribute__((ext_vector_type(8)));

#define NXC 128
#define NYC 64
#define NUC 32
#define NDC 32
#define TSTEPS 4096
#define NTHREADS 1024
#define NWAVES 32

// padded LDS leading dims (stride mod 64 banks != 0)
#define LDP  132   // P
#define LDT  132   // T1 (AP) / CtP
#define LDPC 66    // PC (aliases back half of T1)
#define LDSA 130   // S augmented [S | I], logical 64x128
#define LDL  68    // L
#define LDA  132   // A staged in LDS
#define LDC  68    // C staged in LDS

// D[M, N] = (INIT_IDENT ? I : 0) + op(A)[M,K] * op(B)[K,N] ; optional store (I - D)
// Each wave owns a 16x32 region: one A-fragment feeds two WMMAs per k-step.
// Fragment loads are software-pipelined one k-step ahead of the consuming WMMA.
template<int M, int N, int K, bool TA, bool TB, bool INIT_IDENT, bool EYE_MINUS>
__device__ __forceinline__ void wmma_mm(const float* A, int lda,
                                        const float* B, int ldb,
                                        float* Dst, int ldd,
                                        int wid, int lane)
{
    const int half = lane >> 4;          // 0..1
    const int l16  = lane & 15;          // 0..15
    constexpr int regN   = N / 32;       // 32-wide regions across N
    constexpr int numReg = (M / 16) * regN;
    for (int reg = wid; reg < numReg; reg += NWAVES) {   // wid is wave-scalar
        const int tm  = (reg / regN) * 16;
        const int tn  = (reg % regN) * 32;
        const int am  = tm + l16;
        const int bn0 = tn + l16;
        const int bn1 = tn + 16 + l16;
        v8f c0, c1;
#pragma unroll
        for (int v = 0; v < 8; ++v) {
            const int row = tm + v + 8 * half;
            c0[v] = INIT_IDENT ? ((row == bn0) ? 1.0f : 0.0f) : 0.0f;
            c1[v] = INIT_IDENT ? ((row == bn1) ? 1.0f : 0.0f) : 0.0f;
        }
        // prologue: load k-step 0 fragments
        v2f a, b0, b1;
        {
            const int ka = 2 * half;
            a[0]  = TA ? A[(ka + 0) * lda + am]  : A[am * lda + (ka + 0)];
            a[1]  = TA ? A[(ka + 1) * lda + am]  : A[am * lda + (ka + 1)];
            b0[0] = TB ? B[bn0 * ldb + (ka + 0)] : B[(ka + 0) * ldb + bn0];
            b0[1] = TB ? B[bn0 * ldb + (ka + 1)] : B[(ka + 1) * ldb + bn0];
            b1[0] = TB ? B[bn1 * ldb + (ka + 0)] : B[(ka + 0) * ldb + bn1];
            b1[1] = TB ? B[bn1 * ldb + (ka + 1)] : B[(ka + 1) * ldb + bn1];
        }
        for (int k0 = 0; k0 < K - 4; k0 += 4) {
            const int ka = k0 + 4 + 2 * half;   // next k-step
            v2f an, b0n, b1n;
            an[0]  = TA ? A[(ka + 0) * lda + am]  : A[am * lda + (ka + 0)];
            an[1]  = TA ? A[(ka + 1) * lda + am]  : A[am * lda + (ka + 1)];
            b0n[0] = TB ? B[bn0 * ldb + (ka + 0)] : B[(ka + 0) * ldb + bn0];
            b0n[1] = TB ? B[bn0 * ldb + (ka + 1)] : B[(ka + 1) * ldb + bn0];
            b1n[0] = TB ? B[bn1 * ldb + (ka + 0)] : B[(ka + 0) * ldb + bn1];
            b1n[1] = TB ? B[bn1 * ldb + (ka + 1)] : B[(ka + 1) * ldb + bn1];
            c0 = __builtin_amdgcn_wmma_f32_16x16x4_f32(false, a, false, b0,
                                                       (short)0, c0, false, false);
            c1 = __builtin_amdgcn_wmma_f32_16x16x4_f32(false, a, false, b1,
                                                       (short)0, c1, false, false);
            a = an; b0 = b0n; b1 = b1n;
        }
        c0 = __builtin_amdgcn_wmma_f32_16x16x4_f32(false, a, false, b0,
                                                   (short)0, c0, false, false);
        c1 = __builtin_amdgcn_wmma_f32_16x16x4_f32(false, a, false, b1,
                                                   (short)0, c1, false, false);
#pragma unroll
        for (int v = 0; v < 8; ++v) {
            const int row = tm + v + 8 * half;
            float v0 = c0[v], v1 = c1[v];
            if (EYE_MINUS) {
                v0 = ((row == bn0) ? 1.0f : 0.0f) - v0;
                v1 = ((row == bn1) ? 1.0f : 0.0f) - v1;
            }
            Dst[row * ldd + bn0] = v0;
            Dst[row * ldd + bn1] = v1;
        }
    }
}

__global__ __launch_bounds__(NTHREADS, 1)
void kalman_persistent_kernel(const float* __restrict__ Ym,  // [T,1,NY]
                              const float* __restrict__ U,   // [T,1,NU]
                              const float* __restrict__ Dd,  // [T,1,ND]
                              const float* __restrict__ Ag,  // [NX,NX]
                              const float* __restrict__ Bu,  // [NU,NX]
                              const float* __restrict__ Bd,  // [ND,NX]
                              const float* __restrict__ Cg,  // [NX,NY]
                              float* __restrict__ out)       // [NX+1]
{
    __shared__ float P[NXC * LDP];        // covariance, persistent      (67584 B)
    __shared__ float T1[NXC * LDT];       // AP; later CtP | PC          (67584 B)
    __shared__ float As[NXC * LDA];       // A staged                    (67584 B)
    __shared__ float Cs[NXC * LDC];       // C staged                    (34816 B)
    __shared__ float Saug[NYC * LDSA];    // [S | I] Gauss-Jordan        (33280 B)
    __shared__ float Lm[NXC * LDL];       // Kalman gain, persistent     (34816 B)
    __shared__ float xv[NXC], xp[NXC], ev[NYC], fac[NYC], row_buf[2 * NYC];

    float* CtP = T1;                      // 64 x 128, ld LDT  (64*132 = 8448 floats)
    float* PC  = T1 + NYC * LDT;          // 128 x 64, ld LDPC (128*66 = 8448 floats)

    const int tid  = threadIdx.x;
    const int lane = tid & 31;
    const int wid  = __builtin_amdgcn_readfirstlane(tid >> 5);

    // stage A, C into LDS; init x=0, P=I, L=0
    for (int i = tid; i < NXC * NXC; i += NTHREADS)
        As[(i >> 7) * LDA + (i & 127)] = Ag[i];
    for (int i = tid; i < NXC * NYC; i += NTHREADS)
        Cs[(i >> 6) * LDC + (i & 63)] = Cg[i];
    for (int i = tid; i < NXC * LDP; i += NTHREADS) {
        int r = i / LDP, c = i % LDP;
        P[i] = (r == c) ? 1.0f : 0.0f;
    }
    for (int i = tid; i < NXC * LDL; i += NTHREADS) Lm[i] = 0.0f;
    if (tid < NXC) xv[tid] = 0.0f;
    __syncthreads();

    for (int t = 0; t < TSTEPS; ++t) {
        const float* ym = Ym + t * NYC;
        const float* u  = U  + t * NUC;
        const float* d  = Dd + t * NDC;

        // prefetch next step's streaming inputs (global_prefetch_b8)
        if (t + 1 < TSTEPS) {
            if (tid < 4)            __builtin_prefetch(Ym + (t + 1) * NYC + tid * 16, 0, 0);
            else if (tid < 6)       __builtin_prefetch(U  + (t + 1) * NUC + (tid - 4) * 16, 0, 0);
            else if (tid < 8)       __builtin_prefetch(Dd + (t + 1) * NDC + (tid - 6) * 16, 0, 0);
        }

        // V1: xp = x@A + u@Bu + d@Bd
        if (tid < NXC) {
            float s = 0.0f;
            for (int i = 0; i < NXC; ++i) s += xv[i] * As[i * LDA + tid];
            for (int i = 0; i < NUC; ++i) s += u[i] * Bu[i * NXC + tid];
            for (int i = 0; i < NDC; ++i) s += d[i] * Bd[i * NXC + tid];
            xp[tid] = s;
        }
        __syncthreads();

        // V2: e = ym - xp@C
        if (tid < NYC) {
            float s = 0.0f;
            for (int i = 0; i < NXC; ++i) s += xp[i] * Cs[i * LDC + tid];
            ev[tid] = ym[tid] - s;
        }
        __syncthreads();

        // M1: T1 = A @ P           (128x128x128)
        wmma_mm<NXC, NXC, NXC, false, false, false, false>(As, LDA, P, LDP, T1, LDT, wid, lane);
        __syncthreads();
        // M2: P = T1 @ A^T + I     (128x128x128)
        wmma_mm<NXC, NXC, NXC, false, true, true, false>(T1, LDT, As, LDA, P, LDP, wid, lane);
        __syncthreads();

        // V3: x = xp + e @ L^T   (PREVIOUS gain, as in reference)
        if (tid < NXC) {
            float s = xp[tid];
            for (int j = 0; j < NYC; ++j) s += ev[j] * Lm[tid * LDL + j];
            xv[tid] = s;
        }
        __syncthreads();

        // M3: CtP = C^T @ P        (64x128x128)  [overwrites T1 region]
        wmma_mm<NYC, NXC, NXC, true, false, false, false>(Cs, LDC, P, LDP, CtP, LDT, wid, lane);
        __syncthreads();
        // M4: PC = P @ C           (128x64x128)
        wmma_mm<NXC, NYC, NXC, false, false, false, false>(P, LDP, Cs, LDC, PC, LDPC, wid, lane);
        // M5: Saug[:, :64] = I + CtP @ C   (64x64x128)   (R = I)
        wmma_mm<NYC, NYC, NXC, false, false, true, false>(CtP, LDT, Cs, LDC, Saug, LDSA, wid, lane);
        // Saug[:, 64:128] = I
        for (int i = tid; i < NYC * NYC; i += NTHREADS) {
            int r = i >> 6, c = i & 63;
            Saug[r * LDSA + NYC + c] = (r == c) ? 1.0f : 0.0f;
        }
        __syncthreads();

        // M6: Gauss-Jordan -> Saug[:, 64:128] = inv(S)   (2 barriers per pivot)
        for (int p = 0; p < NYC; ++p) {
            // epoch A: reads only -> side buffers
            if (tid < 2 * NYC) {
                row_buf[tid] = Saug[p * LDSA + tid] * (1.0f / Saug[p * LDSA + p]);
            } else if (tid < 3 * NYC) {
                fac[tid - 2 * NYC] = Saug[(tid - 2 * NYC) * LDSA + p];
            }
            __syncthreads();
            // epoch B: writes only, each thread touches its own elements
            for (int i = tid; i < NYC * 2 * NYC; i += NTHREADS) {
                int r = i >> 7, c = i & 127;
                float cur = Saug[r * LDSA + c];
                Saug[r * LDSA + c] = (r == p) ? row_buf[c]
                                              : cur - fac[r] * row_buf[c];
            }
            __syncthreads();
        }

        // M7: L = PC @ inv(S)      (128x64x64)
        wmma_mm<NXC, NYC, NYC, false, false, false, false>(PC, LDPC, Saug + NYC, LDSA, Lm, LDL, wid, lane);
        __syncthreads();
        // M8: P = I - L @ CtP      (128x128x64)
        wmma_mm<NXC, NXC, NYC, false, false, false, true>(Lm, LDL, CtP, LDT, P, LDP, wid, lane);
        __syncthreads();
    }

    if (tid < NXC) out[tid] = xv[tid];
    if (tid == NXC) out[NXC] = 0.0f;   // reg_error
}

extern "C" void kernel_launch(void* const* d_in, const int* in_sizes, int n_in,
                              void* d_out, int out_size, void* d_ws, size_t ws_size,
                              hipStream_t stream) {
    (void)in_sizes; (void)n_in; (void)out_size; (void)d_ws; (void)ws_size;
    const float* Ym = (const float*)d_in[0];
    const float* U  = (const float*)d_in[1];
    const float* D  = (const float*)d_in[2];
    const float* A  = (const float*)d_in[3];
    const float* Bu = (const float*)d_in[4];
    const float* Bd = (const float*)d_in[5];
    const float* C  = (const float*)d_in[6];
    float* out = (float*)d_out;
    kalman_persistent_kernel<<<1, NTHREADS, 0, stream>>>(Ym, U, D, A, Bu, Bd, C, out);
}